// Intrablock_1477468750193
// MI455X (gfx1250) — compile-verified
//
#include <hip/hip_runtime.h>
#include <hip/hip_bf16.h>

// ---------------------------------------------------------------------------
// Problem constants (from reference): B=8, N=256, D=128, E=4096
// ---------------------------------------------------------------------------
#define KB   8
#define KN   256
#define KD   128
#define KE   4096
#define YSTR 132                      // LDS row stride (floats) -> conflict-free
#define OUT_H_ELEMS (67108864ULL)     // B*N*N*D
#define NUM_WG      (KB * KN)         // 2048 (b,i) blocks

typedef __attribute__((ext_vector_type(2))) float v2f;
typedef __attribute__((ext_vector_type(8))) float v8f;

// ---------------------------------------------------------------------------
// Kernel 0: edge filter -> effective mask, masked-pair count
// ---------------------------------------------------------------------------
__global__ __launch_bounds__(256) void prep_kernel(const int* __restrict__ eidx,
                                                   const unsigned char* __restrict__ mask_l,
                                                   unsigned char* __restrict__ maskEff,
                                                   float* __restrict__ cntp)
{
    __shared__ unsigned char hit[KN];
    __shared__ int cb[KB];
    int t = threadIdx.x;
    hit[t] = 0;
    __syncthreads();
    for (int e = t; e < KE; e += 256) {
        int s = eidx[e];
        int d = eidx[KE + e];
        if (d - s == 3) { hit[s] = 1; hit[d] = 1; }   // benign same-value races
    }
    __syncthreads();
    for (int n = t; n < KB * KN; n += 256) {
        int j = n & (KN - 1);
        maskEff[n] = (mask_l[n] != 0 && hit[j] == 0) ? 1 : 0;
    }
    __syncthreads();
    if (t < KB) {
        int c = 0;
        for (int j = 0; j < KN; ++j) c += (int)maskEff[t * KN + j];
        cb[t] = c;
    }
    __syncthreads();
    if (t == 0) {
        long long tot = 0;
        for (int bb = 0; bb < KB; ++bb) {
            long long c = cb[bb];
            tot += c * c - c;       // pairs with i != j
        }
        *cntp = (float)tot;
    }
}

// ---------------------------------------------------------------------------
// Kernel 1: per-(b,i) GEMM  h[j,e] = sum_d (x[b,i,d]*x[b,j,d]) * W[e,d] + bias[e]
// 256 threads = 8 waves; wave w owns e-tile [16w,16w+16); 16 j-tiles of 16.
// V_WMMA_F32_16X16X4_F32, K=128 -> 32 wmma per tile.
// Also writes pair_mask row and per-block masked (sum, sumsq) partials.
// ---------------------------------------------------------------------------
__global__ __launch_bounds__(256) void pair_gemm_kernel(const float* __restrict__ x,
                                                        const float* __restrict__ W,
                                                        const float* __restrict__ bias,
                                                        const unsigned char* __restrict__ maskEff,
                                                        float* __restrict__ out,
                                                        float* __restrict__ part)
{
    __shared__ __align__(16) float xi[KD];
    __shared__ unsigned char mrow[KN];
    __shared__ __align__(16) float Y[16 * YSTR];
    __shared__ float lsum[256];
    __shared__ float lss[256];

    const int wg = blockIdx.x;
    const int i  = wg & (KN - 1);
    const int b  = wg >> 8;
    const int t  = threadIdx.x;

    // Stage x_i (128 floats) and this batch's mask row.
    if (t < KD / 4) {
        ((float4*)xi)[t] = ((const float4*)x)[(size_t)(b * KN + i) * (KD / 4) + t];
    }
    mrow[t] = maskEff[b * KN + t];
    __syncthreads();
    const int mi = mrow[i];

    // pair_mask output row (float 0/1), at offset B*N*N*D.
    {
        float pm = (mi && mrow[t] && (t != i)) ? 1.0f : 0.0f;
        out[OUT_H_ELEMS + (size_t)wg * KN + t] = pm;
    }

    if (!mi) {                               // whole row masked: no h needed
        part[(size_t)wg * 256 + t] = 0.0f;   // keep reduction deterministic
        return;                              // block-uniform exit (EXEC stays full)
    }

    const int w  = t >> 5;        // wave id -> e-tile
    const int l  = t & 31;
    const int ln = l & 15;
    const int lh = l >> 4;
    const int e  = 16 * w + ln;   // output channel owned by this lane
    const float be = bias[e];

    // B-matrix (W) fragments, register-resident: lane l -> N=ln (row e of W),
    // K = vgprComponent + 2*lh within each 4-wide k-step.
    v2f wf[32];
#pragma unroll
    for (int ks = 0; ks < 32; ++ks) {
        wf[ks] = *(const v2f*)(W + (size_t)e * KD + 4 * ks + 2 * lh);
    }

    float s = 0.0f, ss = 0.0f;
    const size_t outbase = (size_t)wg * KN * KD;

    for (int jt = 0; jt < 16; ++jt) {
        const int j0 = jt * 16;
        __syncthreads();                    // previous Y fully consumed
        // Stage Y[j,d] = x[b,j0+j,d] * xi[d] : 16 rows x 128 cols, 2 float4/thread
#pragma unroll
        for (int q = 0; q < 2; ++q) {
            int idx  = t + 256 * q;         // 0..511
            int row  = idx >> 5;
            int c4   = idx & 31;
            float4 xv = ((const float4*)x)[(size_t)(b * KN + j0 + row) * (KD / 4) + c4];
            float4 iv = ((const float4*)xi)[c4];
            float4 yv;
            yv.x = xv.x * iv.x; yv.y = xv.y * iv.y;
            yv.z = xv.z * iv.z; yv.w = xv.w * iv.w;
            *(float4*)&Y[row * YSTR + 4 * c4] = yv;
        }
        __syncthreads();

        // A: 16x4 tile of Y (lane ln = row M, K = comp + 2*lh); 32 k-steps.
        v8f acc = {};
#pragma unroll
        for (int ks = 0; ks < 32; ++ks) {
            v2f a = *(const v2f*)&Y[ln * YSTR + 4 * ks + 2 * lh];
            acc = __builtin_amdgcn_wmma_f32_16x16x4_f32(
                false, a, false, wf[ks], (short)0, acc, false, false);
        }

        // C/D layout: VGPR r, lanes 0-15 -> M=r, lanes 16-31 -> M=r+8; N=ln.
#pragma unroll
        for (int r = 0; r < 8; ++r) {
            int j = j0 + r + 8 * lh;
            float hv = acc[r] + be;
            if (mrow[j] && (j != i)) { s += hv; ss += hv * hv; }
            out[outbase + (size_t)j * KD + e] = hv;   // raw h; normalized later
        }
    }

    // Fold the two half-wave lanes of each channel -> per-block partials.
    lsum[t] = s;
    lss[t]  = ss;
    __syncthreads();
    if (t < KD) {
        int ww   = t >> 4;
        int base = 32 * ww + (t & 15);
        part[(size_t)wg * 256 + t]       = lsum[base] + lsum[base + 16];
        part[(size_t)wg * 256 + KD + t]  = lss[base]  + lss[base + 16];
    }
}

// ---------------------------------------------------------------------------
// Kernel 2: fixed-order reduction of 2048 partials -> per-channel scale/shift
// ---------------------------------------------------------------------------
__global__ __launch_bounds__(128) void reduce_kernel(const float* __restrict__ part,
                                                     const float* __restrict__ cntp,
                                                     const float* __restrict__ gamma,
                                                     const float* __restrict__ beta,
                                                     float* __restrict__ stats)
{
    int e = threadIdx.x;            // 128 threads, one channel each
    float S = 0.0f, SS = 0.0f;
    for (int wgi = 0; wgi < NUM_WG; ++wgi) {
        S  += part[(size_t)wgi * 256 + e];
        SS += part[(size_t)wgi * 256 + KD + e];
    }
    float cnt = *cntp;
    if (cnt < 1.0f) cnt = 1.0f;
    float mean = S / cnt;
    float var  = SS / cnt - mean * mean;    // biased, matches torch BN
    if (var < 0.0f) var = 0.0f;
    float inv = rsqrtf(var + 1e-5f);
    float sc  = gamma[e] * inv;
    stats[e]      = sc;
    stats[KD + e] = beta[e] - mean * sc;
}

// ---------------------------------------------------------------------------
// Kernel 3: in-place normalize + leaky-ReLU + mask, float4 vectorized
// ---------------------------------------------------------------------------
__global__ __launch_bounds__(256) void finalize_kernel(float* __restrict__ out,
                                                       const float* __restrict__ stats,
                                                       const unsigned char* __restrict__ maskEff)
{
    __shared__ float sc[KD];
    __shared__ float sh[KD];
    int t = threadIdx.x;
    if (t < KD) { sc[t] = stats[t]; sh[t] = stats[KD + t]; }
    __syncthreads();

    size_t idx4 = (size_t)blockIdx.x * 256 + t;   // over B*N*N*(D/4)
    int    c4   = (int)(idx4 & 31);               // channel group
    size_t pair = idx4 >> 5;                      // b*N*N + i*N + j
    int j  = (int)(pair & 255);
    int ii = (int)((pair >> 8) & 255);
    int bb = (int)(pair >> 16);
    bool m = maskEff[bb * KN + ii] && maskEff[bb * KN + j] && (ii != j);

    float4* o4 = (float4*)out;
    if (m) {
        float4 hv = o4[idx4];
        int e = 4 * c4;
        float4 r;
        float v;
        v = hv.x * sc[e + 0] + sh[e + 0]; r.x = (v > 0.0f) ? v : 0.01f * v;
        v = hv.y * sc[e + 1] + sh[e + 1]; r.y = (v > 0.0f) ? v : 0.01f * v;
        v = hv.z * sc[e + 2] + sh[e + 2]; r.z = (v > 0.0f) ? v : 0.01f * v;
        v = hv.w * sc[e + 3] + sh[e + 3]; r.w = (v > 0.0f) ? v : 0.01f * v;
        o4[idx4] = r;
    } else {
        float4 z; z.x = 0.0f; z.y = 0.0f; z.z = 0.0f; z.w = 0.0f;
        o4[idx4] = z;                     // covers masked rows (d_out is poisoned)
    }
}

// ---------------------------------------------------------------------------
// Host entry
// ---------------------------------------------------------------------------
extern "C" void kernel_launch(void* const* d_in, const int* in_sizes, int n_in,
                              void* d_out, int out_size, void* d_ws, size_t ws_size,
                              hipStream_t stream)
{
    const float*         x      = (const float*)d_in[0];
    const int*           eidx   = (const int*)d_in[1];
    const unsigned char* mask_l = (const unsigned char*)d_in[2];  // numpy bool = 1 byte
    const float*         W      = (const float*)d_in[3];
    const float*         bias   = (const float*)d_in[4];
    const float*         gamma  = (const float*)d_in[5];
    const float*         beta   = (const float*)d_in[6];
    float*               out    = (float*)d_out;

    char* ws = (char*)d_ws;
    unsigned char* maskEff = (unsigned char*)ws;        // 2048 B
    float* cntp  = (float*)(ws + 2048);                 // 4 B
    float* stats = (float*)(ws + 2560);                 // 256 floats (scale, shift)
    float* part  = (float*)(ws + 4096);                 // 2048 * 256 floats ~= 2 MB

    prep_kernel<<<1, 256, 0, stream>>>(eidx, mask_l, maskEff, cntp);
    pair_gemm_kernel<<<NUM_WG, 256, 0, stream>>>(x, W, bias, maskEff, out, part);
    reduce_kernel<<<1, 128, 0, stream>>>(part, cntp, gamma, beta, stats);
    finalize_kernel<<<(int)(OUT_H_ELEMS / 4 / 256), 256, 0, stream>>>(out, stats, maskEff);
}